// HouseholderFlow_49082886259470
// MI455X (gfx1250) — compile-verified
//
#include <hip/hip_runtime.h>

// HouseholderFlow for MI455X (gfx1250, wave32).
//   phase 1: v_new = v @ W^T + b   via V_WMMA_F32_16X16X4_F32 (exact fp32),
//            tiles staged with GLOBAL_LOAD_ASYNC_TO_LDS_B128 (ASYNCcnt),
//            double-buffered LDS for copy/compute overlap (constant buffer ids
//            so fragment reads stay on the DS path, not FLAT).
//   phase 2: x_out = x - 2 v_new (v_new . x)/||v_new||^2 ; logdet = 0
//
// d_out layout: [x_out (B*D) | v_new (B*D) | logdet (1)]  (fp32)

typedef __attribute__((ext_vector_type(2))) float v2f;
typedef __attribute__((ext_vector_type(8))) float v8f;

#define NB 8192      // batch rows
#define ND 2048      // feature dim
#define BM 128       // C-tile rows per block
#define BN 128       // C-tile cols per block
#define BK 32        // k-depth per LDS stage
#define LDSS 36      // padded LDS row stride in floats (16B-aligned, bank-friendly)

// One per-lane 16B async DMA: global -> LDS, no VGPR staging, tracked by ASYNCcnt.
__device__ __forceinline__ void async_b128(unsigned lds_off, const float* g) {
    asm volatile("global_load_async_to_lds_b128 %0, %1, off"
                 :: "v"(lds_off), "v"(g) : "memory");
}

// 8 WMMAs per 4-deep k-step over the wave's 32x64 C sub-tile.
__device__ __forceinline__ void tile_mma(const float* ar, const float* br,
                                         v8f acc[2][4]) {
    #pragma unroll
    for (int ks = 0; ks < BK / 4; ++ks) {
        v2f af0 = *(const v2f*)(ar + ks * 4);
        v2f af1 = *(const v2f*)(ar + 16 * LDSS + ks * 4);
        #pragma unroll
        for (int nt = 0; nt < 4; ++nt) {
            v2f bf = *(const v2f*)(br + nt * 16 * LDSS + ks * 4);
            acc[0][nt] = __builtin_amdgcn_wmma_f32_16x16x4_f32(
                false, af0, false, bf, (short)0, acc[0][nt], false, false);
            acc[1][nt] = __builtin_amdgcn_wmma_f32_16x16x4_f32(
                false, af1, false, bf, (short)0, acc[1][nt], false, false);
        }
    }
}

__global__ __launch_bounds__(256)
void hh_gemm_vnew_kernel(const float* __restrict__ V,
                         const float* __restrict__ W,
                         const float* __restrict__ bias,
                         float* __restrict__ vnew)
{
    __shared__ float As[2][BM * LDSS];
    __shared__ float Bs[2][BN * LDSS];

    const int tid  = threadIdx.x;      // 0..255, 8 waves
    const int lane = tid & 31;
    const int wv   = tid >> 5;         // wave id 0..7
    const int wm   = wv >> 1;          // 0..3 -> 32-row slice of C-tile
    const int wn   = wv & 1;           // 0..1 -> 64-col slice of C-tile
    const int half = lane >> 4;        // lane group (K select)
    const int l16  = lane & 15;        // M (A) / N (B) index within 16

    const int m0 = blockIdx.y * BM;
    const int n0 = blockIdx.x * BN;

    // cooperative staging: thread t copies 16 floats of row (t>>1), cols (t&1)*16..+15
    const int lrow = tid >> 1;
    const int lcol = (tid & 1) << 4;

    const float* Arow = V + (size_t)(m0 + lrow) * ND + lcol;
    const float* Brow = W + (size_t)(n0 + lrow) * ND + lcol;

    // LDS byte offsets for this thread's staging strip (flat addr low 32 bits = LDS offset)
    const unsigned aoff0 = (unsigned)(size_t)&As[0][lrow * LDSS + lcol];
    const unsigned aoff1 = (unsigned)(size_t)&As[1][lrow * LDSS + lcol];
    const unsigned boff0 = (unsigned)(size_t)&Bs[0][lrow * LDSS + lcol];
    const unsigned boff1 = (unsigned)(size_t)&Bs[1][lrow * LDSS + lcol];

    // per-lane fragment element offsets (A: row = M, B: row = N; +2*half selects K pair)
    const int afo = (wm * 32 + l16) * LDSS + (half << 1);
    const int bfo = (wn * 64 + l16) * LDSS + (half << 1);

    v8f acc[2][4] = {};

#define ISSUE_TILE(ao, bo, kbase)                                     \
    do {                                                              \
        _Pragma("unroll")                                             \
        for (int q = 0; q < 4; ++q) {                                 \
            async_b128((ao) + q * 16, Arow + (kbase) + q * 4);        \
            async_b128((bo) + q * 16, Brow + (kbase) + q * 4);        \
        }                                                             \
    } while (0)

    // prologue: stage tile 0 into buffer 0 (8 async b128 per thread)
    ISSUE_TILE(aoff0, boff0, 0);

    const int NT = ND / BK;   // 64 k-tiles (even)
    for (int kt = 0; kt < NT; kt += 2) {
        // ---- tile kt from buffer 0; prefetch tile kt+1 -> buffer 1
        ISSUE_TILE(aoff1, boff1, (kt + 1) * BK);      // kt+1 < NT always
        asm volatile("s_wait_asynccnt 0x8" ::: "memory");  // buf0 transfers done
        __syncthreads();
        tile_mma(&As[0][afo], &Bs[0][bfo], acc);
        __syncthreads();                              // buf0 free for refill

        // ---- tile kt+1 from buffer 1; prefetch tile kt+2 -> buffer 0
        if (kt + 2 < NT) {
            ISSUE_TILE(aoff0, boff0, (kt + 2) * BK);
            asm volatile("s_wait_asynccnt 0x8" ::: "memory");
        } else {
            asm volatile("s_wait_asynccnt 0x0" ::: "memory");
        }
        __syncthreads();
        tile_mma(&As[1][afo], &Bs[1][bfo], acc);
        __syncthreads();
    }
#undef ISSUE_TILE

    // epilogue: bias add + store (C/D layout: lane -> N, vgpr r + 8*half -> M)
    #pragma unroll
    for (int nt = 0; nt < 4; ++nt) {
        const int n  = n0 + wn * 64 + nt * 16 + l16;
        const float bn = bias[n];
        #pragma unroll
        for (int mt = 0; mt < 2; ++mt) {
            const int mbase = m0 + wm * 32 + mt * 16 + half * 8;
            #pragma unroll
            for (int r = 0; r < 8; ++r) {
                vnew[(size_t)(mbase + r) * ND + n] = acc[mt][nt][r] + bn;
            }
        }
    }
}

__global__ __launch_bounds__(256)
void hh_reflect_kernel(const float* __restrict__ X,
                       const float* __restrict__ vnew,
                       float* __restrict__ xout,
                       float* __restrict__ logdet)
{
    const int row = blockIdx.x;
    const int tid = threadIdx.x;

    const float4* vr = (const float4*)(vnew + (size_t)row * ND);
    const float4* xr = (const float4*)(X    + (size_t)row * ND);
    const float4 va = vr[tid], vb = vr[tid + 256];
    const float4 xa = xr[tid], xb = xr[tid + 256];

    float svv = va.x*va.x + va.y*va.y + va.z*va.z + va.w*va.w
              + vb.x*vb.x + vb.y*vb.y + vb.z*vb.z + vb.w*vb.w;
    float svx = va.x*xa.x + va.y*xa.y + va.z*xa.z + va.w*xa.w
              + vb.x*xb.x + vb.y*xb.y + vb.z*xb.z + vb.w*xb.w;

    // wave32 butterfly reduction
    #pragma unroll
    for (int off = 16; off > 0; off >>= 1) {
        svv += __shfl_xor(svv, off, 32);
        svx += __shfl_xor(svx, off, 32);
    }

    __shared__ float rvv[8], rvx[8];
    if ((tid & 31) == 0) { rvv[tid >> 5] = svv; rvx[tid >> 5] = svx; }
    __syncthreads();
    float tvv = 0.f, tvx = 0.f;
    #pragma unroll
    for (int i = 0; i < 8; ++i) { tvv += rvv[i]; tvx += rvx[i]; }

    const float s = 2.0f * (tvx / tvv);
    float4 oa, ob;
    oa.x = xa.x - s * va.x;  oa.y = xa.y - s * va.y;
    oa.z = xa.z - s * va.z;  oa.w = xa.w - s * va.w;
    ob.x = xb.x - s * vb.x;  ob.y = xb.y - s * vb.y;
    ob.z = xb.z - s * vb.z;  ob.w = xb.w - s * vb.w;

    float4* orow = (float4*)(xout + (size_t)row * ND);
    orow[tid]       = oa;
    orow[tid + 256] = ob;

    if (row == 0 && tid == 0) *logdet = 0.0f;
}

extern "C" void kernel_launch(void* const* d_in, const int* in_sizes, int n_in,
                              void* d_out, int out_size, void* d_ws, size_t ws_size,
                              hipStream_t stream) {
    (void)in_sizes; (void)n_in; (void)d_ws; (void)ws_size; (void)out_size;
    const float* x = (const float*)d_in[0];
    const float* v = (const float*)d_in[1];
    const float* W = (const float*)d_in[2];
    const float* b = (const float*)d_in[3];

    float* xout   = (float*)d_out;
    float* vnew   = xout + (size_t)NB * ND;
    float* logdet = xout + 2 * (size_t)NB * ND;

    dim3 gg(ND / BN, NB / BM);                    // 16 x 64 blocks
    hh_gemm_vnew_kernel<<<gg, 256, 0, stream>>>(v, W, b, vnew);
    hh_reflect_kernel<<<NB, 256, 0, stream>>>(x, vnew, xout, logdet);
}